// PairwiseGenerativeRetrievalLoss_60198261621391
// MI455X (gfx1250) — compile-verified
//
#include <hip/hip_runtime.h>
#include <stdint.h>

#define TT 24
#define BB 32
#define VV 32000
#define CHUNK4 500            // float4s per chunk = 2000 floats
#define NCHUNK 16             // 16 * 2000 = 32000 = VV exactly
#define BLK 256
#define NWAVE 8               // 256 threads / wave32

// ---------------- CDNA5 async global->LDS path ----------------
#if defined(__has_builtin)
# if __has_builtin(__builtin_amdgcn_global_load_async_to_lds_b128)
#  define USE_ASYNC_BUILTIN 1
# endif
# if __has_builtin(__builtin_amdgcn_s_wait_asynccnt)
#  define USE_WAIT_BUILTIN 1
# endif
#endif

typedef int v4i __attribute__((ext_vector_type(4)));
typedef __attribute__((address_space(1))) v4i g1_v4i;   // global (AS1)
typedef __attribute__((address_space(3))) v4i l3_v4i;   // LDS (AS3)
typedef __attribute__((address_space(3))) void lds_void;

__device__ __forceinline__ void async_ld_b128(void* lds_dst, const void* gsrc) {
#ifdef USE_ASYNC_BUILTIN
  __builtin_amdgcn_global_load_async_to_lds_b128(
      (g1_v4i*)gsrc, (l3_v4i*)lds_dst, 0, 0);
#else
  uint32_t l = (uint32_t)(uintptr_t)(lds_void*)lds_dst;
  asm volatile("global_load_async_to_lds_b128 %0, %1, off"
               :: "v"(l), "v"(gsrc) : "memory");
#endif
}

#ifdef USE_WAIT_BUILTIN
# define WAIT_ASYNCCNT(n) __builtin_amdgcn_s_wait_asynccnt(n)
#else
# define WAIT_ASYNCCNT(n) asm volatile("s_wait_asynccnt %0" :: "i"(n) : "memory")
#endif

// ---------------- Threefry-2x32 (20 rounds) ----------------
__device__ __forceinline__ uint32_t rotl32(uint32_t x, int r) {
  return (x << r) | (x >> (32 - r));
}

__device__ __forceinline__ void threefry2x32(uint32_t k0, uint32_t k1,
                                             uint32_t c0, uint32_t c1,
                                             uint32_t& o0, uint32_t& o1) {
  const uint32_t ks0 = k0, ks1 = k1, ks2 = k0 ^ k1 ^ 0x1BD11BDAu;
  uint32_t x0 = c0 + ks0, x1 = c1 + ks1;
  const int RA[4] = {13, 15, 26, 6};
  const int RB[4] = {17, 29, 16, 24};
#pragma unroll
  for (int i = 0; i < 4; ++i) { x0 += x1; x1 = rotl32(x1, RA[i]); x1 ^= x0; }
  x0 += ks1; x1 += ks2 + 1u;
#pragma unroll
  for (int i = 0; i < 4; ++i) { x0 += x1; x1 = rotl32(x1, RB[i]); x1 ^= x0; }
  x0 += ks2; x1 += ks0 + 2u;
#pragma unroll
  for (int i = 0; i < 4; ++i) { x0 += x1; x1 = rotl32(x1, RA[i]); x1 ^= x0; }
  x0 += ks0; x1 += ks1 + 3u;
#pragma unroll
  for (int i = 0; i < 4; ++i) { x0 += x1; x1 = rotl32(x1, RB[i]); x1 ^= x0; }
  x0 += ks1; x1 += ks2 + 4u;
#pragma unroll
  for (int i = 0; i < 4; ++i) { x0 += x1; x1 = rotl32(x1, RA[i]); x1 ^= x0; }
  x0 += ks2; x1 += ks0 + 5u;
  o0 = x0; o1 = x1;
}

// ---------------- Kernel A: per-step RNG state ----------------
__global__ void pgrl_rng_setup(int* __restrict__ targets,
                               uint32_t* __restrict__ subkeys) {
  const int t = threadIdx.x;
  if (t >= TT) return;
  uint32_t key0, key1;
  threefry2x32(0u, 42u, 0u, (uint32_t)t, key0, key1);      // fold_in(key(42), t)
  uint32_t kt0, kt1, ksp0, ksp1;
  threefry2x32(key0, key1, 0u, 1u, kt0, kt1);              // kt
  threefry2x32(key0, key1, 2u, 3u, ksp0, ksp1);            // ks
  uint32_t r0, r1;
  threefry2x32(kt0, kt1, 0u, 0u, r0, r1);
  const int target = (int)(r0 % 3u);                       // randint(0,3)
  uint32_t s0, s1;
  if (target == 0)      threefry2x32(ksp0, ksp1, 0u, 1u, s0, s1);   // ks1 (p)
  else if (target == 1) threefry2x32(ksp0, ksp1, 2u, 3u, s0, s1);   // ks2 (n)
  else                  threefry2x32(ksp0, ksp1, 4u, 5u, s0, s1);   // ks3 (q)
  targets[t] = target;
  subkeys[2 * t + 0] = s0;
  subkeys[2 * t + 1] = s1;
}

// ---------------- Kernel B: streaming V-reduction ----------------
__global__ void __launch_bounds__(BLK) pgrl_reduce(
    const float* __restrict__ P, const float* __restrict__ N,
    const float* __restrict__ Q, const int* __restrict__ targets,
    const uint32_t* __restrict__ subkeys, float* __restrict__ recs) {
  __shared__ float4 tp[2][CHUNK4];
  __shared__ float4 tn[2][CHUNK4];
  __shared__ float4 tq[2][CHUNK4];
  __shared__ float  rsum[NWAVE][7];

  const int tid = threadIdx.x;
  const int blk = blockIdx.x;
  const int t = blk / BB;
  const int b = blk - t * BB;
  const size_t row = ((size_t)t * BB + b) * (size_t)VV;
  const float4* gp = (const float4*)(P + row);
  const float4* gn = (const float4*)(N + row);
  const float4* gq = (const float4*)(Q + row);
  const int target = targets[t];
  const uint32_t sk0 = subkeys[2 * t], sk1 = subkeys[2 * t + 1];

  // Each thread async-copies the exact LDS words it will later consume:
  // the steady-state pipeline needs only per-wave s_wait_asynccnt, no barriers.
  auto issue = [&](int c, int buf) {
    const float4* sp = gp + c * CHUNK4;
    const float4* sn = gn + c * CHUNK4;
    const float4* sq = gq + c * CHUNK4;
    for (int i = tid; i < CHUNK4; i += BLK) {
      async_ld_b128(&tp[buf][i], sp + i);
      async_ld_b128(&tn[buf][i], sn + i);
      async_ld_b128(&tq[buf][i], sq + i);
    }
  };

  issue(0, 0);

  float s1 = 0.f, s2 = 0.f, s3 = 0.f, s4 = 0.f, s5 = 0.f;
  float bestV = -3.4e38f;
  int bestI = 0;

  for (int c = 0; c < NCHUNK; ++c) {
    const int buf = c & 1;
    if (c + 1 < NCHUNK) {
      issue(c + 1, buf ^ 1);
      WAIT_ASYNCCNT(6);   // 6 async ops/wave/chunk in flight max; oldest 6 retired
    } else {
      WAIT_ASYNCCNT(0);
    }
    for (int i = tid; i < CHUNK4; i += BLK) {
      const float4 p4 = tp[buf][i];
      const float4 n4 = tn[buf][i];
      const float4 q4 = tq[buf][i];
      const int vbase = c * (CHUNK4 * 4) + i * 4;
#pragma unroll
      for (int e = 0; e < 4; ++e) {
        const float p = (e == 0) ? p4.x : (e == 1) ? p4.y : (e == 2) ? p4.z : p4.w;
        const float n = (e == 0) ? n4.x : (e == 1) ? n4.y : (e == 2) ? n4.z : n4.w;
        const float q = (e == 0) ? q4.x : (e == 1) ? q4.y : (e == 2) ? q4.z : q4.w;
        const float pq  = p * q;
        const float lpq = __logf(pq);     // log(p)+log(q) fused into one log
        const float ln  = __logf(n);
        const float a   = pq * (1.0f - n);
        const float nln = n * ln;
        s1 += pq;
        s2 += a;
        s3 += a * lpq;
        s4 += nln;
        s5 += pq * nln;
        // Gumbel-argmax categorical for the selected distribution only
        const float d    = (target == 0) ? p : (target == 1) ? n : q;
        const float logd = (target == 1) ? ln : __logf(d);
        const uint32_t v = (uint32_t)(vbase + e);
        uint32_t r0, r1;
        threefry2x32(sk0, sk1, (uint32_t)b, v, r0, r1);
        float u = fmaxf((float)(r0 >> 8) * 5.9604645e-08f, 1.1754944e-38f);
        const float g = -__logf(-__logf(u));
        const float sc = logd + g;
        if (sc > bestV) { bestV = sc; bestI = (int)v; }
      }
    }
  }

  // wave32 reduction
#pragma unroll
  for (int o = 16; o > 0; o >>= 1) {
    s1 += __shfl_down(s1, o);
    s2 += __shfl_down(s2, o);
    s3 += __shfl_down(s3, o);
    s4 += __shfl_down(s4, o);
    s5 += __shfl_down(s5, o);
    const float ov = __shfl_down(bestV, o);
    const int   oi = __shfl_down(bestI, o);
    if (ov > bestV || (ov == bestV && oi < bestI)) { bestV = ov; bestI = oi; }
  }
  const int wave = tid >> 5, lane = tid & 31;
  if (lane == 0) {
    rsum[wave][0] = s1; rsum[wave][1] = s2; rsum[wave][2] = s3;
    rsum[wave][3] = s4; rsum[wave][4] = s5; rsum[wave][5] = bestV;
    rsum[wave][6] = __int_as_float(bestI);
  }
  __syncthreads();
  if (tid == 0) {
    float S1 = 0.f, S2 = 0.f, S3 = 0.f, S4 = 0.f, S5 = 0.f, bV = -3.4e38f;
    int bI = 0;
#pragma unroll
    for (int w = 0; w < NWAVE; ++w) {
      S1 += rsum[w][0]; S2 += rsum[w][1]; S3 += rsum[w][2];
      S4 += rsum[w][3]; S5 += rsum[w][4];
      const float v = rsum[w][5];
      const int   i = __float_as_int(rsum[w][6]);
      if (v > bV || (v == bV && i < bI)) { bV = v; bI = i; }
    }
    const int tok = bI;
    const float pt = P[row + tok];
    const float nt = N[row + tok];
    const float qt = Q[row + tok];
    float* r = recs + (size_t)blk * 8;
    r[0] = S2;                     // coefficient of L_t = sum(log cur)
    r[1] = S3 + S1 * S4 - S5;      // cur-independent part of ml_t
    r[2] = pt; r[3] = nt; r[4] = qt;
  }
}

// ---------------- Kernel C: sequential t-scan + reverse fold ----------------
__global__ void pgrl_finalize(const float* __restrict__ recs,
                              const int* __restrict__ targets,
                              float* __restrict__ out) {
  const int b = threadIdx.x;
  if (b >= BB) return;
  float ml[TT], mu[TT];
  float L = 0.f;
#pragma unroll
  for (int t = 0; t < TT; ++t) {
    const float* r = recs + ((size_t)t * BB + b) * 8;
    const float coef = r[0], base = r[1];
    const float pt = r[2], nt = r[3], qt = r[4];
    ml[t] = base + coef * L;                       // uses pre-update cur
    L += __logf(pt) + __logf(nt) + __logf(qt);     // cur *= (p_t, n_t, q_t)
    const int tg = targets[t];
    mu[t] = (tg == 0) ? nt * qt : (tg == 1) ? pt * qt : pt * nt;
  }
  float res = ml[TT - 1];
#pragma unroll
  for (int t = TT - 2; t >= 0; --t) res = res * mu[t] + ml[t];
  out[b] = res;
}

// ---------------- launch ----------------
extern "C" void kernel_launch(void* const* d_in, const int* in_sizes, int n_in,
                              void* d_out, int out_size, void* d_ws, size_t ws_size,
                              hipStream_t stream) {
  const float* P = (const float*)d_in[0];   // posdoc_probs [T,B,V] f32
  const float* N = (const float*)d_in[1];   // negdoc_probs
  const float* Q = (const float*)d_in[2];   // query_probs
  float* out = (float*)d_out;               // res [B] f32

  char* ws = (char*)d_ws;
  int*      targets = (int*)(ws);                 //  96 B
  uint32_t* subkeys = (uint32_t*)(ws + 256);      // 192 B
  float*    recs    = (float*)(ws + 1024);        // T*B*8 floats = 24576 B

  pgrl_rng_setup<<<1, 32, 0, stream>>>(targets, subkeys);
  pgrl_reduce<<<TT * BB, BLK, 0, stream>>>(P, N, Q, targets, subkeys, recs);
  pgrl_finalize<<<1, 32, 0, stream>>>(recs, targets, out);
}